// TrainedLoraModel_67585605369954
// MI455X (gfx1250) — compile-verified
//
#include <hip/hip_runtime.h>
#include <hip/hip_bf16.h>
#include <math.h>

typedef __attribute__((ext_vector_type(2))) float v2f;
typedef __attribute__((ext_vector_type(8))) float v8f;

#define Bc    8
#define Hc    128
#define Wc    128
#define Ec    768
#define Kc    16
#define HWc   16384
#define TOPK  5
#define CHUNK 64
#define NCHUNK 12          // 768 / 64
#define APAD  68           // padded LDS row stride (floats): 4-bank shift per row
#define NEG_BIG -3.402823466e38f

// ---------------------------------------------------------------------------
// K0: L2-normalize rows of length 768 (desc: 128 rows, query: 8 rows)
// ---------------------------------------------------------------------------
__global__ __launch_bounds__(256) void normalize_rows(const float* __restrict__ in,
                                                      float* __restrict__ out) {
  const int row = blockIdx.x;
  const int t = threadIdx.x;
  const float* x = in + (size_t)row * Ec;
  float s = 0.f;
  for (int i = t; i < Ec; i += 256) { float v = x[i]; s += v * v; }
  __shared__ float red[256];
  red[t] = s;
  __syncthreads();
  for (int off = 128; off; off >>= 1) {
    if (t < off) red[t] += red[t + off];
    __syncthreads();
  }
  const float inv = 1.0f / fmaxf(sqrtf(red[0]), 1e-12f);
  for (int i = t; i < Ec; i += 256) out[(size_t)row * Ec + i] = x[i] * inv;
}

// ---------------------------------------------------------------------------
// K1: fused single-pass over the map:
//   scores[b,k,pix] = (x . descn[k]) / max(||x||, 1e-12)   via V_WMMA_F32_16X16X4_F32
//   qdotv[b,pix]    = (x . qn)       / max(||x||, 1e-12)
//   mask zeroed.
// grid = B*256 blocks of 128 threads; each of 4 waves owns a 16-pixel tile.
// ---------------------------------------------------------------------------
__global__ __launch_bounds__(128) void score_gemm(const float* __restrict__ map,
                                                  const float* __restrict__ descn,
                                                  const float* __restrict__ qn,
                                                  float* __restrict__ scores,
                                                  float* __restrict__ qdotv,
                                                  unsigned char* __restrict__ mask) {
  __shared__ float sA[4][16 * APAD];   // per-wave A chunk (16 x 64, padded)
  __shared__ float sD[16 * APAD];      // shared desc chunk (16 x 64, padded)
  __shared__ float sPS[4][32][8];      // sum-of-squares partials
  __shared__ float sPQ[4][32][8];      // q-dot partials
  __shared__ float sInv[4][16];        // 1/rownorm

  const int tid  = threadIdx.x;
  const int wave = tid >> 5;
  const int lane = tid & 31;
  const int b    = blockIdx.x >> 8;          // 256 blocks per batch
  const int t64  = blockIdx.x & 255;         // 64-pixel group
  const int pix0 = t64 * 64 + wave * 16;     // this wave's 16-row tile

  const float* mapB = map   + ((size_t)b * HWc + pix0) * Ec;
  const float* dscB = descn + (size_t)b * Kc * Ec;
  const float* qB   = qn    + (size_t)b * Ec;

  const int lh  = lane >> 4;        // 0/1 half of wave
  const int ll  = lane & 15;
  const int cc  = ll << 2;          // column group this lane loads (fixed)

  float psum[8], pq[8];
#pragma unroll
  for (int i = 0; i < 8; ++i) { psum[i] = 0.f; pq[i] = 0.f; }

  v8f acc = {};

  for (int ch = 0; ch < NCHUNK; ++ch) {
    const int k0 = ch * CHUNK;
    __syncthreads();

    // prefetch next K-chunk of this lane's rows (global_prefetch_b8)
    if (ch + 1 < NCHUNK)
      __builtin_prefetch(mapB + (size_t)lh * Ec + k0 + CHUNK + cc, 0, 3);

    // --- stage A chunk (16x64) for this wave; coalesced float4 loads ---
    const float4 q4 = *(const float4*)(qB + k0 + cc);
#pragma unroll
    for (int i = 0; i < 8; ++i) {
      const int r = lh + 2 * i;    // lane's row for slot i (fixed pattern)
      float4 v = *(const float4*)(mapB + (size_t)r * Ec + k0 + cc);
      *(float4*)&sA[wave][r * APAD + cc] = v;
      psum[i] += v.x * v.x + v.y * v.y + v.z * v.z + v.w * v.w;
      pq[i]   += v.x * q4.x + v.y * q4.y + v.z * q4.z + v.w * q4.w;
    }

    // --- stage desc chunk (16x64) cooperatively (all 128 threads) ---
#pragma unroll
    for (int i = 0; i < 2; ++i) {
      const int f = tid + (i << 7);
      const int r = f >> 4;
      const int c2 = (f & 15) << 2;
      *(float4*)&sD[r * APAD + c2] =
          *(const float4*)(dscB + (size_t)r * Ec + k0 + c2);
    }
    __syncthreads();

    // --- 16 WMMA k-steps (K=4 each) ---
    // A 16x4 f32 layout: lanes 0-15 -> M=lane, K=0,1 ; lanes 16-31 -> M=lane-16, K=2,3
    // B 4x16 f32 layout mirrors with N in place of M.
    const int koff = lh << 1;  // 0 or 2
#pragma unroll
    for (int kk = 0; kk < 16; ++kk) {
      v2f a  = *(const v2f*)&sA[wave][ll * APAD + (kk << 2) + koff];
      v2f bt = *(const v2f*)&sD[ll * APAD + (kk << 2) + koff];
      acc = __builtin_amdgcn_wmma_f32_16x16x4_f32(
          /*neg_a=*/false, a, /*neg_b=*/false, bt,
          /*c_mod=*/(short)0, acc, /*reuse_a=*/false, /*reuse_b=*/false);
    }
  }

  // --- deterministic reduction of rownorm^2 and q-dot ---
#pragma unroll
  for (int i = 0; i < 8; ++i) { sPS[wave][lane][i] = psum[i]; sPQ[wave][lane][i] = pq[i]; }
  __syncthreads();
  if (lane < 16) {
    const int h = lane & 1, i = lane >> 1;  // row 'lane' = h + 2*i pattern owner group
    float ss = 0.f, qq = 0.f;
    for (int j = 0; j < 16; ++j) {
      ss += sPS[wave][h * 16 + j][i];
      qq += sPQ[wave][h * 16 + j][i];
    }
    const float inv = 1.0f / fmaxf(sqrtf(ss), 1e-12f);
    sInv[wave][lane] = inv;
    const size_t pix = (size_t)b * HWc + pix0 + lane;
    qdotv[pix] = qq * inv;      // == fm . q_normalized
    mask[pix]  = 0;             // zero the mask plane (set later by NMS)
  }
  __syncthreads();

  // --- scale + store scores: C layout: lane holds col n=lane&15, rows v + 8*(lane>=16) ---
  const int nd    = ll;                 // descriptor index
  const int mbase = lh << 3;            // 0 or 8
  float* sc = scores + ((size_t)b * Kc + nd) * HWc + pix0;
#pragma unroll
  for (int v = 0; v < 8; ++v) {
    const int m = mbase + v;
    sc[m] = acc[v] * sInv[wave][m];
  }
}

// ---------------------------------------------------------------------------
// K2: greedy top-5 argmax with radius-2 Chebyshev NMS per (b,k) map, then
// write the 3x3-dilated (border-clipped) mask union. One block per (b,k).
// Suppression done on-the-fly against already-selected centers.
// ---------------------------------------------------------------------------
__global__ __launch_bounds__(256) void nms_mask(const float* __restrict__ scores,
                                                unsigned char* __restrict__ mask) {
  const int t = threadIdx.x;
  const int b = blockIdx.x >> 4;   // Kc = 16
  const float* sc = scores + (size_t)blockIdx.x * HWc;
  unsigned char* mk = mask + (size_t)b * HWc;

  __shared__ int   selR[TOPK], selC[TOPK];
  __shared__ float rv[256];
  __shared__ int   ri[256];

  for (int it = 0; it < TOPK; ++it) {
    float best = NEG_BIG;
    int bi = 0;
    for (int i = t; i < HWc; i += 256) {
      const int r = i >> 7, c = i & 127;
      bool sup = false;
      for (int j = 0; j < it; ++j)
        sup |= (abs(r - selR[j]) <= 2) && (abs(c - selC[j]) <= 2);
      const float v = sup ? NEG_BIG : sc[i];
      if (v > best) { best = v; bi = i; }   // strided i ascending -> first max kept
    }
    rv[t] = best; ri[t] = bi;
    __syncthreads();
    for (int off = 128; off; off >>= 1) {
      if (t < off) {
        if (rv[t + off] > rv[t] || (rv[t + off] == rv[t] && ri[t + off] < ri[t])) {
          rv[t] = rv[t + off]; ri[t] = ri[t + off];
        }
      }
      __syncthreads();
    }
    const int idx = ri[0];
    const int r = idx >> 7, c = idx & 127;
    if (t == 0) { selR[it] = r; selC[it] = c; }
    if (t < 9) { // 3x3 dilation with border clipping (all writes are 1: benign race)
      const int rr = min(max(r + t / 3 - 1, 0), Hc - 1);
      const int cc = min(max(c + t % 3 - 1, 0), Wc - 1);
      mk[rr * Wc + cc] = 1;
    }
    __syncthreads();
  }
}

// ---------------------------------------------------------------------------
// K3: value_map = mask ? (fm.q) : 0 ; fused soft-argmax (tau = 0.1).
// One block per batch.
// ---------------------------------------------------------------------------
__global__ __launch_bounds__(256) void valmap_softargmax(const float* __restrict__ qdotv,
                                                         const unsigned char* __restrict__ mask,
                                                         float* __restrict__ out_val,
                                                         float* __restrict__ out_crd) {
  const int b = blockIdx.x;
  const int t = threadIdx.x;
  const float* qd = qdotv + (size_t)b * HWc;
  const unsigned char* mk = mask + (size_t)b * HWc;
  float* ov = out_val + (size_t)b * HWc;

  __shared__ float r0[256], r1[256], r2[256];

  // pass 1: write value map, find max
  float mx = NEG_BIG;
  for (int i = t; i < HWc; i += 256) {
    const float v = mk[i] ? qd[i] : 0.0f;
    ov[i] = v;
    mx = fmaxf(mx, v);
  }
  r0[t] = mx;
  __syncthreads();
  for (int off = 128; off; off >>= 1) {
    if (t < off) r0[t] = fmaxf(r0[t], r0[t + off]);
    __syncthreads();
  }
  mx = r0[0];
  __syncthreads();

  // pass 2: exp sums (recompute v; data is L2-hot)
  float se = 0.f, sr = 0.f, scl = 0.f;
  for (int i = t; i < HWc; i += 256) {
    const float v = mk[i] ? qd[i] : 0.0f;
    const float p = expf((v - mx) * 10.0f);   // 1/tau
    se += p;
    sr  += p * (float)(i >> 7);
    scl += p * (float)(i & 127);
  }
  r0[t] = se; r1[t] = sr; r2[t] = scl;
  __syncthreads();
  for (int off = 128; off; off >>= 1) {
    if (t < off) { r0[t] += r0[t + off]; r1[t] += r1[t + off]; r2[t] += r2[t + off]; }
    __syncthreads();
  }
  if (t == 0) {
    out_crd[b * 2 + 0] = r1[0] / r0[0];
    out_crd[b * 2 + 1] = r2[0] / r0[0];
  }
}

// ---------------------------------------------------------------------------
extern "C" void kernel_launch(void* const* d_in, const int* in_sizes, int n_in,
                              void* d_out, int out_size, void* d_ws, size_t ws_size,
                              hipStream_t stream) {
  (void)in_sizes; (void)n_in; (void)out_size; (void)ws_size;
  const float* map   = (const float*)d_in[0];  // (B,H,W,E)
  const float* desc  = (const float*)d_in[1];  // (B,K,E)
  const float* query = (const float*)d_in[2];  // (B,E)
  // d_in[3] = gt_coords: unused by reference outputs

  float* out_val = (float*)d_out;              // (B,H,W,1) flat
  float* out_crd = out_val + (size_t)Bc * HWc; // (B,2)

  float* ws     = (float*)d_ws;
  float* descn  = ws;                               // B*K*E
  float* qn     = descn + (size_t)Bc * Kc * Ec;     // B*E
  float* scores = qn + (size_t)Bc * Ec;             // B*K*HW
  float* qdotv  = scores + (size_t)Bc * Kc * HWc;   // B*HW
  unsigned char* mask = (unsigned char*)(qdotv + (size_t)Bc * HWc); // B*HW bytes

  normalize_rows<<<Bc * Kc, 256, 0, stream>>>(desc, descn);
  normalize_rows<<<Bc, 256, 0, stream>>>(query, qn);
  score_gemm<<<Bc * (HWc / 64), 128, 0, stream>>>(map, descn, qn, scores, qdotv, mask);
  nms_mask<<<Bc * Kc, 256, 0, stream>>>(scores, mask);
  valmap_softargmax<<<Bc, 256, 0, stream>>>(qdotv, mask, out_val, out_crd);
}